// CompositeLoss_33509334843682
// MI455X (gfx1250) — compile-verified
//
#include <hip/hip_runtime.h>
#include <math.h>

// ---------------------------------------------------------------------------
// CompositeLoss for MI455X (gfx1250): fused LNCC (9^3 box filter via
// V_WMMA_F32_16X16X4_F32 banded matmuls + z register ring) + deterministic
// two-stage reductions for the jac/vel/liquid terms.
// ---------------------------------------------------------------------------

#define TILE   16
#define PADR   4
#define REGW   24            // TILE + 2*PADR
#define WIN    9
#define NVOX   128           // D = H = W
#define NINTH  (1.0f / 9.0f)
#define EPSV   1e-8f

#define PRODSZ (REGW * REGW)   // 576
#define MIDSZ  (REGW * TILE)   // 384
#define OUTSZ  (TILE * TILE)   // 256

typedef __attribute__((ext_vector_type(2))) float v2f;
typedef __attribute__((ext_vector_type(8))) float v8f;

// ---- gfx1250 async global->LDS copy (ASYNCcnt path), guarded -------------
#if defined(__HIP_DEVICE_COMPILE__) && \
    __has_builtin(__builtin_amdgcn_global_load_async_to_lds_b32)
#define HAVE_ASYNC_COPY 1
#else
#define HAVE_ASYNC_COPY 0
#endif

#if HAVE_ASYNC_COPY
typedef __attribute__((address_space(1))) int g_as_int;
typedef __attribute__((address_space(3))) int l_as_int;
#endif

__device__ __forceinline__ void async_copy_f32(const float* g, float* l) {
#if HAVE_ASYNC_COPY
  __builtin_amdgcn_global_load_async_to_lds_b32(
      (g_as_int*)(void*)(const_cast<float*>(g)),
      (l_as_int*)(void*)l, 0, 0);
#else
  *l = *g;
#endif
}

__device__ __forceinline__ void wait_async() {
#if HAVE_ASYNC_COPY
#if __has_builtin(__builtin_amdgcn_s_wait_asynccnt)
  __builtin_amdgcn_s_wait_asynccnt(0);
#else
  asm volatile("s_wait_asynccnt 0" ::: "memory");
#endif
#endif
}

// ---- f32 WMMA: D(16x16) = A(16x4) * B(4x16) + C --------------------------
__device__ __forceinline__ v8f wmma4(v2f a, v2f b, v8f c) {
  // 8-arg pattern: (neg_a, A, neg_b, B, c_mod, C, reuse_a, reuse_b)
  return __builtin_amdgcn_wmma_f32_16x16x4_f32(false, a, false, b,
                                               (short)0, c, false, false);
}

// ---- fixed-order block reduction (deterministic, no atomics) -------------
__device__ __forceinline__ float blockReduce256(float v, float* sh) {
  const int lane = threadIdx.x & 31;
  const int wv   = threadIdx.x >> 5;
#pragma unroll
  for (int o = 16; o > 0; o >>= 1) v += __shfl_down(v, o, 32);
  if (lane == 0) sh[wv] = v;
  __syncthreads();
  float r = (threadIdx.x < 8) ? sh[threadIdx.x] : 0.f;
  if (wv == 0) {
#pragma unroll
    for (int o = 16; o > 0; o >>= 1) r += __shfl_down(r, o, 32);
  }
  __syncthreads();
  return r;   // valid in thread 0
}

__device__ __forceinline__ int reflect128(int i) {
  if (i < 0) i = -i;
  if (i > NVOX - 1) i = 2 * (NVOX - 1) - i;
  return i;
}

// ---------------------------------------------------------------------------
// LNCC kernel: block = 16x16 xy output tile, sweep z with a 9-deep ring.
// 5 moment fields (f, w, f^2, w^2, fw) are box-filtered in x then y via
// banded-matrix WMMAs on the f32 matrix pipe, then combined per voxel.
// All LDS tables live in single flat arrays so fragment fetches lower to
// ds_load_b64 off one base (no flat-address pointer selects).
// ---------------------------------------------------------------------------
__global__ __launch_bounds__(256)
void lncc_kernel(const float* __restrict__ fixedv,
                 const float* __restrict__ warpedv,
                 float* __restrict__ partials) {
  __shared__ float sProd[5 * PRODSZ];   // f, w, f*f, w*w, f*w  (24x24 each)
  __shared__ float sMid[5 * MIDSZ];     // x-filtered [yrow 0..23][x 0..15]
  __shared__ float sOutS[5 * OUTSZ];    // xy-filtered slice
  __shared__ float sRed[8];

  const int tid  = threadIdx.x;
  const int lane = tid & 31;
  const int wave = tid >> 5;
  const int x0 = blockIdx.x * TILE;
  const int y0 = blockIdx.y * TILE;
  const int bb = blockIdx.z;

  const float* fb = fixedv  + (size_t)bb * NVOX * NVOX * NVOX;
  const float* wb = warpedv + (size_t)bb * NVOX * NVOX * NVOX;

  // Banded box-filter fragments, hoisted out of the z loop. For output
  // index j = lane&15 and K-slot k: band = 1/9 iff j <= k <= j+8. The same
  // values serve as the x-filter B fragment and the y-filter A fragment.
  const int n    = lane & 15;
  const int koff = (lane < 16) ? 0 : 2;   // per-half-wave K slot (A/B layout)
  v2f band[6];
#pragma unroll
  for (int kb = 0; kb < 6; ++kb) {
    const int k0 = kb * 4 + koff;
    band[kb].x = (k0     >= n && k0     <= n + 8) ? NINTH : 0.f;
    band[kb].y = (k0 + 1 >= n && k0 + 1 <= n + 8) ? NINTH : 0.f;
  }

  float ring[5][WIN];
#pragma unroll
  for (int q = 0; q < 5; ++q)
#pragma unroll
    for (int i = 0; i < WIN; ++i) ring[q][i] = 0.f;

  float lsum = 0.f;

  for (int t = -PADR; t < NVOX + PADR; ++t) {
    const int zm = reflect128(t);
    const float* fslice = fb + (size_t)zm * NVOX * NVOX;
    const float* wslice = wb + (size_t)zm * NVOX * NVOX;

    __syncthreads();
    // --- async fill of the 24x24 halo region (reflect-padded) -------------
    for (int e = tid; e < PRODSZ; e += 256) {
      const int yy = e / REGW;
      const int xx = e - yy * REGW;
      const int gy = reflect128(y0 - PADR + yy);
      const int gx = reflect128(x0 - PADR + xx);
      async_copy_f32(fslice + gy * NVOX + gx, &sProd[e]);
      async_copy_f32(wslice + gy * NVOX + gx, &sProd[PRODSZ + e]);
    }
    wait_async();
    __syncthreads();

    for (int e = tid; e < PRODSZ; e += 256) {
      const float f = sProd[e], w = sProd[PRODSZ + e];
      sProd[2 * PRODSZ + e] = f * f;
      sProd[3 * PRODSZ + e] = w * w;
      sProd[4 * PRODSZ + e] = f * w;
    }
    __syncthreads();

    // --- x-filter: Mid(24x16) = In(24x24) * Sx(24x16), WMMA K-chunks of 4 --
    // 10 wave-uniform tile tasks: 5 products x 2 row-chunks (rows 0..15,16..23)
    for (int task = wave; task < 10; task += 8) {
      const int p  = task >> 1;
      const int rc = task & 1;
      int mrow = rc * 16 + n;
      if (mrow > REGW - 1) mrow = REGW - 1;        // junk rows, discarded
      const int aoff = p * PRODSZ + mrow * REGW + koff;  // 8B-aligned
      v2f af[6];
#pragma unroll
      for (int kb = 0; kb < 6; ++kb) {             // clause of ds_load_b64
        af[kb].x = sProd[aoff + kb * 4];
        af[kb].y = sProd[aoff + kb * 4 + 1];
      }
      v8f acc = {};
#pragma unroll
      for (int kb = 0; kb < 6; ++kb) acc = wmma4(af[kb], band[kb], acc);
#pragma unroll
      for (int r = 0; r < 8; ++r) {
        const int row = rc * 16 + r + ((lane < 16) ? 0 : 8);
        if (row < REGW) sMid[p * MIDSZ + row * TILE + n] = acc[r];
      }
    }
    __syncthreads();

    // --- y-filter: Out(16x16) = Sy(16x24) * Mid(24x16) ---------------------
    if (wave < 5) {
      const int p = wave;
      const int boff = p * MIDSZ + koff * TILE + n;
      v2f bf[6];
#pragma unroll
      for (int kb = 0; kb < 6; ++kb) {             // 2addr ds loads
        bf[kb].x = sMid[boff + kb * 4 * TILE];
        bf[kb].y = sMid[boff + kb * 4 * TILE + TILE];
      }
      v8f acc = {};
#pragma unroll
      for (int kb = 0; kb < 6; ++kb) acc = wmma4(band[kb], bf[kb], acc);
#pragma unroll
      for (int r = 0; r < 8; ++r) {
        const int m = r + ((lane < 16) ? 0 : 8);
        sOutS[p * OUTSZ + m * TILE + n] = acc[r];
      }
    }
    __syncthreads();

    // --- z ring update + pointwise LNCC ------------------------------------
    const int tx = tid & 15, ty = tid >> 4;
#pragma unroll
    for (int q = 0; q < 5; ++q) {
#pragma unroll
      for (int i = 0; i < WIN - 1; ++i) ring[q][i] = ring[q][i + 1];
      ring[q][WIN - 1] = sOutS[q * OUTSZ + ty * TILE + tx];
    }

    if (t >= PADR) {
      float s[5];
#pragma unroll
      for (int q = 0; q < 5; ++q) {
        float a = 0.f;
#pragma unroll
        for (int i = 0; i < WIN; ++i) a += ring[q][i];
        s[q] = a * NINTH;
      }
      const float mu_f = s[0], mu_w = s[1];
      const float sff = fmaxf(s[2] - mu_f * mu_f, EPSV);
      const float sww = fmaxf(s[3] - mu_w * mu_w, EPSV);
      const float sfw = s[4] - mu_f * mu_w;
      const float den = fmaxf(sqrtf(fmaxf(sff * sww, EPSV * EPSV)), EPSV);
      float r = sfw / den;
      if (r != r) r = 0.f;                          // nan_to_num
      r = fminf(fmaxf(r, -10.f), 10.f);
      lsum += r;
    }
  }

  const float bsum = blockReduce256(lsum, sRed);
  if (tid == 0)
    partials[(blockIdx.z * 8 + blockIdx.y) * 8 + blockIdx.x] = bsum;
}

// ---------------------------------------------------------------------------
// Streaming sum-of-squares (bias 0 for velocity, 1 for jacobian).
// ---------------------------------------------------------------------------
__global__ __launch_bounds__(256)
void sumsq_kernel(const float4* __restrict__ in, float* __restrict__ partials,
                  int n4, float bias) {
  __shared__ float sh[8];
  const int stride = gridDim.x * 256;
  float acc = 0.f;
  for (int i = blockIdx.x * 256 + threadIdx.x; i < n4; i += stride) {
    const float4 v = in[i];
    __builtin_prefetch(in + i + stride, 0, 0);     // global_prefetch_b8
    const float a = v.x - bias, b = v.y - bias, c = v.z - bias, d = v.w - bias;
    acc = fmaf(a, a, acc); acc = fmaf(b, b, acc);
    acc = fmaf(c, c, acc); acc = fmaf(d, d, acc);
  }
  const float r = blockReduce256(acc, sh);
  if (threadIdx.x == 0) partials[blockIdx.x] = r;
}

__global__ __launch_bounds__(256)
void liquid_kernel(const float* __restrict__ in, float* __restrict__ out2,
                   int n) {
  __shared__ float sh[8];
  float sq = 0.f, rl = 0.f;
  for (int i = threadIdx.x; i < n; i += 256) {
    const float v = in[i];
    sq = fmaf(v, v, sq);
    rl += fmaxf(fabsf(v) - 1.f, 0.f);
  }
  const float s = blockReduce256(sq, sh);
  if (threadIdx.x == 0) out2[0] = s;
  __syncthreads();
  const float r = blockReduce256(rl, sh);
  if (threadIdx.x == 0) out2[1] = r;
}

// ---------------------------------------------------------------------------
// Finalize: fixed-order reduction of all partials, emit the 5 outputs.
// ---------------------------------------------------------------------------
__global__ __launch_bounds__(256)
void finalize_kernel(const float* __restrict__ velP,
                     const float* __restrict__ jacP,
                     const float* __restrict__ lnccP,
                     const float* __restrict__ liqP,
                     float* __restrict__ out) {
  __shared__ float sh[8];
  const int tid = threadIdx.x;

  float a = 0.f;
  for (int i = tid; i < 1024; i += 256) a += velP[i];
  const float velSum = blockReduce256(a, sh);
  __syncthreads();

  a = 0.f;
  for (int i = tid; i < 512; i += 256) a += jacP[i];
  const float jacSum = blockReduce256(a, sh);
  __syncthreads();

  a = (tid < 128) ? lnccP[tid] : 0.f;
  const float lnccSum = blockReduce256(a, sh);

  if (tid == 0) {
    const float nL  = 4194304.0f;   // 2*128^3
    const float nJ  = 4000752.0f;   // 2*126^3
    const float nVl = 12582912.0f;  // 2*3*128^3
    const float nQ  = 2048.0f;      // 2*1024
    const float sim = -(lnccSum / nL);                        // LAM_SIM = 1
    const float jac = jacSum / nJ;                            // LAM_JAC = 1
    const float vel = 0.01f * (velSum / nVl);                 // LAM_VEL
    const float liq = 0.001f * (liqP[0] / nQ + liqP[1] / nQ); // LAM_LIQ
    out[0] = sim; out[1] = jac; out[2] = vel; out[3] = liq;
    out[4] = sim + jac + vel + liq;
  }
}

// ---------------------------------------------------------------------------
extern "C" void kernel_launch(void* const* d_in, const int* in_sizes, int n_in,
                              void* d_out, int out_size, void* d_ws,
                              size_t ws_size, hipStream_t stream) {
  const float* fixedv = (const float*)d_in[0];
  const float* warpedv = (const float*)d_in[1];
  const float* velv = (const float*)d_in[2];
  const float* jacv = (const float*)d_in[3];
  const float* liqv = (const float*)d_in[4];
  float* out = (float*)d_out;
  float* ws = (float*)d_ws;

  float* velP  = ws;          // 1024 partials
  float* jacP  = ws + 1024;   // 512 partials
  float* lnccP = ws + 1536;   // 128 partials
  float* liqP  = ws + 1664;   // 2 partials

  const int n4_vel = (2 * 3 * 128 * 128 * 128) / 4;   // 3,145,728
  const int n4_jac = (2 * 126 * 126 * 126) / 4;       // 1,000,188

  sumsq_kernel<<<1024, 256, 0, stream>>>((const float4*)velv, velP, n4_vel, 0.f);
  sumsq_kernel<<<512, 256, 0, stream>>>((const float4*)jacv, jacP, n4_jac, 1.f);
  liquid_kernel<<<1, 256, 0, stream>>>(liqv, liqP, 2 * 1024);

  dim3 grid(8, 8, 2);
  lncc_kernel<<<grid, 256, 0, stream>>>(fixedv, warpedv, lnccP);

  finalize_kernel<<<1, 256, 0, stream>>>(velP, jacP, lnccP, liqP, out);
}